// CC_67912022885165
// MI455X (gfx1250) — compile-verified
//
#include <hip/hip_runtime.h>
#include <math.h>

// Criss-cross attention (CCNet) for MI455X / gfx1250, fp32 WMMA path.
// B=8, CQ=64, CV=512, H=W=128.

#define B_   8
#define CQ_  64
#define CV_  512
#define H_   128
#define W_   128
#define HW_  (H_ * W_)

typedef __attribute__((ext_vector_type(2))) float v2f;
typedef __attribute__((ext_vector_type(8))) float v8f;

// D(16x16,f32) = A(16x4,f32) * B(4x16,f32) + C
__device__ __forceinline__ v8f wmma_f32(v2f a, v2f b, v8f c) {
    return __builtin_amdgcn_wmma_f32_16x16x4_f32(false, a, false, b,
                                                 (short)0, c, false, false);
}

// Q/K staging rows padded: 144 % 32 == 16 so the two half-waves (K and K+2
// rows) land on disjoint LDS bank ranges for A/B reads.
#define LDSTRIDE 144
#define PSTRIDE  17    // per-wave P tile [j][m], odd stride: conflict-free
#define VSTRIDE  129   // V tile [c][j]

// ---- E strip: 16 rows x 128 cols of E = Qs^T * Ks for one wave -------------
// A(m,k)=qs[k-row][row0+m], B(k,n)=ks[k-row][16t+n]. f32 WMMA register map:
// A/B vgpr0 holds K={0 | 2} for lanes {0-15 | 16-31}, vgpr1 holds K={1 | 3}.
__device__ __forceinline__ void energy_strip(const float* __restrict__ qs,
                                             const float* __restrict__ ks,
                                             int row0, int ln, int hi,
                                             v8f acc[8]) {
    for (int c0 = 0; c0 < CQ_; c0 += 4) {
        v2f a;
        a.x = qs[(c0 + 2 * hi)     * LDSTRIDE + row0 + ln];
        a.y = qs[(c0 + 2 * hi + 1) * LDSTRIDE + row0 + ln];
#pragma unroll
        for (int t = 0; t < 8; ++t) {
            v2f b;
            b.x = ks[(c0 + 2 * hi)     * LDSTRIDE + t * 16 + ln];
            b.y = ks[(c0 + 2 * hi + 1) * LDSTRIDE + t * 16 + ln];
            acc[t] = wmma_f32(a, b, acc[t]);
        }
    }
}

// ---- per-row max / sum(exp) over a 16x128 strip ----------------------------
// C/D layout: element r of lane L is E[row0 + r + 8*(L/16)][16t + (L%16)].
// Butterfly xor {1,2,4,8} reduces across the 16 lanes of each half-wave.
__device__ __forceinline__ void strip_stats(const v8f acc[8],
                                            float rmax[8], float rsum[8]) {
#pragma unroll
    for (int r = 0; r < 8; ++r) rmax[r] = -INFINITY;
#pragma unroll
    for (int t = 0; t < 8; ++t)
#pragma unroll
        for (int r = 0; r < 8; ++r) rmax[r] = fmaxf(rmax[r], acc[t][r]);
#pragma unroll
    for (int r = 0; r < 8; ++r) {
        float m = rmax[r];
        m = fmaxf(m, __shfl_xor(m, 1, 32));
        m = fmaxf(m, __shfl_xor(m, 2, 32));
        m = fmaxf(m, __shfl_xor(m, 4, 32));
        m = fmaxf(m, __shfl_xor(m, 8, 32));
        rmax[r] = m;
    }
#pragma unroll
    for (int r = 0; r < 8; ++r) rsum[r] = 0.0f;
#pragma unroll
    for (int t = 0; t < 8; ++t)
#pragma unroll
        for (int r = 0; r < 8; ++r) rsum[r] += __expf(acc[t][r] - rmax[r]);
#pragma unroll
    for (int r = 0; r < 8; ++r) {
        float s = rsum[r];
        s += __shfl_xor(s, 1, 32);
        s += __shfl_xor(s, 2, 32);
        s += __shfl_xor(s, 4, 32);
        s += __shfl_xor(s, 8, 32);
        rsum[r] = s;
    }
}

// ================= Kernel 1: column-attention softmax stats =================
__global__ __launch_bounds__(256, 1)
void cc_col_stats(const float* __restrict__ Q, const float* __restrict__ K,
                  float* __restrict__ mH, float* __restrict__ sH) {
    __shared__ float qs[CQ_ * LDSTRIDE];
    __shared__ float ks[CQ_ * LDSTRIDE];
    const int b = blockIdx.x / W_, w = blockIdx.x % W_;
    const int tid = threadIdx.x;
    const float* Qb = Q + (size_t)b * CQ_ * HW_ + w;
    const float* Kb = K + (size_t)b * CQ_ * HW_ + w;
    for (int e = tid; e < CQ_ * H_; e += 256) {
        int c = e >> 7, x = e & 127;
        qs[c * LDSTRIDE + x] = Qb[c * HW_ + x * W_];
        ks[c * LDSTRIDE + x] = Kb[c * HW_ + x * W_];
    }
    __syncthreads();

    const int wave = tid >> 5, lane = tid & 31, ln = lane & 15, hi = lane >> 4;
    const int i0 = wave * 16;
    v8f zero = {};
    v8f acc[8];
#pragma unroll
    for (int t = 0; t < 8; ++t) acc[t] = zero;
    energy_strip(qs, ks, i0, ln, hi, acc);

    // diag mask: i == j  <=>  t == wave && ln == r + 8*hi
#pragma unroll
    for (int t = 0; t < 8; ++t) {
        if (t == wave) {
#pragma unroll
            for (int r = 0; r < 8; ++r)
                if (ln == r + 8 * hi) acc[t][r] = -INFINITY;
        }
    }

    float rmax[8], rsum[8];
    strip_stats(acc, rmax, rsum);
    if (ln == 0) {
#pragma unroll
        for (int r = 0; r < 8; ++r) {
            int i = i0 + r + 8 * hi;
            int idx = (b * H_ + i) * W_ + w;
            mH[idx] = rmax[r];
            sH[idx] = rsum[r];
        }
    }
}

// ================= Kernel 2: row-attention softmax stats ====================
__global__ __launch_bounds__(256, 1)
void cc_row_stats(const float* __restrict__ Q, const float* __restrict__ K,
                  float* __restrict__ mW, float* __restrict__ sW) {
    __shared__ float qs[CQ_ * LDSTRIDE];
    __shared__ float ks[CQ_ * LDSTRIDE];
    const int b = blockIdx.x / H_, i = blockIdx.x % H_;
    const int tid = threadIdx.x;
    const float* Qb = Q + (size_t)b * CQ_ * HW_ + i * W_;
    const float* Kb = K + (size_t)b * CQ_ * HW_ + i * W_;
    for (int e = tid; e < CQ_ * W_; e += 256) {
        int c = e >> 7, x = e & 127;
        qs[c * LDSTRIDE + x] = Qb[c * HW_ + x];
        ks[c * LDSTRIDE + x] = Kb[c * HW_ + x];
    }
    __syncthreads();

    const int wave = tid >> 5, lane = tid & 31, ln = lane & 15, hi = lane >> 4;
    const int w0 = wave * 16;
    v8f zero = {};
    v8f acc[8];
#pragma unroll
    for (int t = 0; t < 8; ++t) acc[t] = zero;
    energy_strip(qs, ks, w0, ln, hi, acc);   // E_W(w, j), no mask

    float rmax[8], rsum[8];
    strip_stats(acc, rmax, rsum);
    if (ln == 0) {
#pragma unroll
        for (int r = 0; r < 8; ++r) {
            int w = w0 + r + 8 * hi;
            int idx = (b * H_ + i) * W_ + w;
            mW[idx] = rmax[r];
            sW[idx] = rsum[r];
        }
    }
}

// ================= Kernel 3: merge softmax stats ============================
__global__ __launch_bounds__(256)
void cc_combine(float* __restrict__ mH, float* __restrict__ sH,
                const float* __restrict__ mW, const float* __restrict__ sW,
                int n) {
    int idx = blockIdx.x * 256 + threadIdx.x;
    if (idx >= n) return;
    float m1 = mH[idx], m2 = mW[idx];
    float m = fmaxf(m1, m2);
    float s = sH[idx] * __expf(m1 - m) + sW[idx] * __expf(m2 - m);
    mH[idx] = m;
    sH[idx] = s;
}

// ========== Kernel 4: column pass -> out = att_H^T applied to V ============
__global__ __launch_bounds__(256, 1)
void cc_col_out(const float* __restrict__ Q, const float* __restrict__ K,
                const float* __restrict__ V, const float* __restrict__ Ms,
                const float* __restrict__ Ss, float* __restrict__ out) {
    __shared__ float qk[2 * CQ_ * LDSTRIDE];   // phase1: Q|K, phase2: P tiles
    __shared__ float vs[16 * VSTRIDE];
    float* qs = qk;
    float* ks = qk + CQ_ * LDSTRIDE;

    const int b = blockIdx.x / W_, w = blockIdx.x % W_;
    const int tid = threadIdx.x;
    const float* Qb = Q + (size_t)b * CQ_ * HW_ + w;
    const float* Kb = K + (size_t)b * CQ_ * HW_ + w;
    for (int e = tid; e < CQ_ * H_; e += 256) {
        int c = e >> 7, x = e & 127;
        qs[c * LDSTRIDE + x] = Qb[c * HW_ + x * W_];
        ks[c * LDSTRIDE + x] = Kb[c * HW_ + x * W_];
    }
    __syncthreads();

    const int wave = tid >> 5, lane = tid & 31, ln = lane & 15, hi = lane >> 4;
    const int i0 = wave * 16;
    v8f zero = {};
    v8f acc[8];
#pragma unroll
    for (int t = 0; t < 8; ++t) acc[t] = zero;
    energy_strip(qs, ks, i0, ln, hi, acc);
#pragma unroll
    for (int t = 0; t < 8; ++t) {
        if (t == wave) {
#pragma unroll
            for (int r = 0; r < 8; ++r)
                if (ln == r + 8 * hi) acc[t][r] = -INFINITY;
        }
    }

    // P = exp(E - M) / S  (in place)
    float Mr[8], Si[8];
#pragma unroll
    for (int r = 0; r < 8; ++r) {
        int idx = (b * H_ + i0 + r + 8 * hi) * W_ + w;
        Mr[r] = Ms[idx];
        Si[r] = 1.0f / Ss[idx];
    }
#pragma unroll
    for (int t = 0; t < 8; ++t)
#pragma unroll
        for (int r = 0; r < 8; ++r)
            acc[t][r] = __expf(acc[t][r] - Mr[r]) * Si[r];

    __syncthreads();   // all waves done reading qs/ks; safe to overwrite with P
    float* pl = qk + wave * (H_ * PSTRIDE);   // per-wave P, layout [j][m]
#pragma unroll
    for (int t = 0; t < 8; ++t)
#pragma unroll
        for (int r = 0; r < 8; ++r)
            pl[(t * 16 + ln) * PSTRIDE + r + 8 * hi] = acc[t][r];

    // Out^T(i, c) = sum_j P(i,j) * V(c,j); loop over c-tiles, V tile in LDS.
    const float* Vb = V + (size_t)b * CV_ * HW_ + w;
    for (int cb = 0; cb < CV_; cb += 16) {
        __syncthreads();
        for (int e = tid; e < 16 * H_; e += 256) {
            int c = e >> 7, j = e & 127;
            vs[c * VSTRIDE + j] = Vb[(size_t)(cb + c) * HW_ + j * W_];
        }
        __syncthreads();
        v8f o = zero;
        for (int j0 = 0; j0 < H_; j0 += 4) {
            v2f a, bb;
            a.x  = pl[(j0 + 2 * hi)     * PSTRIDE + ln];   // A = P (m=i,k=j)
            a.y  = pl[(j0 + 2 * hi + 1) * PSTRIDE + ln];
            bb.x = vs[ln * VSTRIDE + j0 + 2 * hi];          // B = V^T (k=j,n=c)
            bb.y = vs[ln * VSTRIDE + j0 + 2 * hi + 1];
            o = wmma_f32(a, bb, o);
        }
        // D: M = i-local = r+8*hi, N = c-local = ln
#pragma unroll
        for (int r = 0; r < 8; ++r) {
            size_t oidx = ((size_t)(b * CV_ + cb + ln) * H_ + i0 + r + 8 * hi) * W_ + w;
            out[oidx] = o[r];
        }
    }
}

// ========== Kernel 5: row pass -> out += att_W applied to V =================
__global__ __launch_bounds__(256, 1)
void cc_row_out(const float* __restrict__ Q, const float* __restrict__ K,
                const float* __restrict__ V, const float* __restrict__ Ms,
                const float* __restrict__ Ss, float* __restrict__ out) {
    __shared__ float qk[2 * CQ_ * LDSTRIDE];
    __shared__ float vs[16 * VSTRIDE];
    float* qs = qk;
    float* ks = qk + CQ_ * LDSTRIDE;

    const int b = blockIdx.x / H_, i = blockIdx.x % H_;
    const int tid = threadIdx.x;
    const float* Qb = Q + (size_t)b * CQ_ * HW_ + i * W_;
    const float* Kb = K + (size_t)b * CQ_ * HW_ + i * W_;
    for (int e = tid; e < CQ_ * W_; e += 256) {
        int c = e >> 7, x = e & 127;
        qs[c * LDSTRIDE + x] = Qb[c * HW_ + x];
        ks[c * LDSTRIDE + x] = Kb[c * HW_ + x];
    }
    __syncthreads();

    const int wave = tid >> 5, lane = tid & 31, ln = lane & 15, hi = lane >> 4;
    const int w0 = wave * 16;
    v8f zero = {};
    v8f acc[8];
#pragma unroll
    for (int t = 0; t < 8; ++t) acc[t] = zero;
    energy_strip(qs, ks, w0, ln, hi, acc);   // E_W(w, j)

    float Mr[8], Si[8];
#pragma unroll
    for (int r = 0; r < 8; ++r) {
        int idx = (b * H_ + i) * W_ + w0 + r + 8 * hi;
        Mr[r] = Ms[idx];
        Si[r] = 1.0f / Ss[idx];
    }
#pragma unroll
    for (int t = 0; t < 8; ++t)
#pragma unroll
        for (int r = 0; r < 8; ++r)
            acc[t][r] = __expf(acc[t][r] - Mr[r]) * Si[r];

    __syncthreads();
    float* pl = qk + wave * (W_ * PSTRIDE);   // per-wave P_W, layout [j][w]
#pragma unroll
    for (int t = 0; t < 8; ++t)
#pragma unroll
        for (int r = 0; r < 8; ++r)
            pl[(t * 16 + ln) * PSTRIDE + r + 8 * hi] = acc[t][r];

    // Out(c, w) = sum_j V(c,j) * P_W(w,j); A = V tile, B = P^T.
    const float* Vb = V + (size_t)b * CV_ * HW_ + i * W_;
    for (int cb = 0; cb < CV_; cb += 16) {
        __syncthreads();
        for (int e = tid; e < 16 * W_; e += 256) {
            int c = e >> 7, j = e & 127;
            vs[c * VSTRIDE + j] = Vb[(size_t)(cb + c) * HW_ + j];
        }
        __syncthreads();
        v8f o = zero;
        for (int j0 = 0; j0 < W_; j0 += 4) {
            v2f a, bb;
            a.x  = vs[ln * VSTRIDE + j0 + 2 * hi];          // A = V (m=c,k=j)
            a.y  = vs[ln * VSTRIDE + j0 + 2 * hi + 1];
            bb.x = pl[(j0 + 2 * hi)     * PSTRIDE + ln];    // B = P^T (k=j,n=w)
            bb.y = pl[(j0 + 2 * hi + 1) * PSTRIDE + ln];
            o = wmma_f32(a, bb, o);
        }
        // D: M = c-local = r+8*hi, N = w-local = ln; lane-contiguous RMW.
#pragma unroll
        for (int r = 0; r < 8; ++r) {
            size_t oidx = ((size_t)(b * CV_ + cb + r + 8 * hi) * H_ + i) * W_ + w0 + ln;
            out[oidx] += o[r];
        }
    }
}

extern "C" void kernel_launch(void* const* d_in, const int* in_sizes, int n_in,
                              void* d_out, int out_size, void* d_ws, size_t ws_size,
                              hipStream_t stream) {
    const float* Q = (const float*)d_in[0];
    const float* K = (const float*)d_in[1];
    const float* V = (const float*)d_in[2];
    float* out = (float*)d_out;

    const int N = B_ * H_ * W_;          // 131072 softmax rows
    float* mH = (float*)d_ws;            // later: merged max M
    float* sH = mH + N;                  // later: merged sum S
    float* mW = sH + N;
    float* sW = mW + N;

    dim3 blk(256);
    cc_col_stats<<<dim3(B_ * W_), blk, 0, stream>>>(Q, K, mH, sH);
    cc_row_stats<<<dim3(B_ * H_), blk, 0, stream>>>(Q, K, mW, sW);
    cc_combine<<<dim3((N + 255) / 256), blk, 0, stream>>>(mH, sH, mW, sW, N);
    cc_col_out<<<dim3(B_ * W_), blk, 0, stream>>>(Q, K, V, mH, sH, out);
    cc_row_out<<<dim3(B_ * H_), blk, 0, stream>>>(Q, K, V, mH, sH, out);
}